// HeteroSAGE_57956288692353
// MI455X (gfx1250) — compile-verified
//
#include <hip/hip_runtime.h>

// ---------------------------------------------------------------------------
// HeteroSAGE (2-layer, 4-relation) for MI455X / gfx1250.
//   D = 128 features. NQ = NP = 100000 nodes, E = 600000 edges per relation.
//   - Segment-mean via L2-resident fp32 scatter atomics (tables fit in 192MB L2)
//   - Fused  h@W_self + (nsum/deg)@W_neigh + bias  with v_wmma_f32_16x16x4_f32
//   - Tile staging via GLOBAL_LOAD_ASYNC_TO_LDS_B128 (ASYNCcnt async path)
// ---------------------------------------------------------------------------

typedef float v2f __attribute__((ext_vector_type(2)));
typedef float v8f __attribute__((ext_vector_type(8)));

#define D 128
#define ROWS 32       // rows per block (two 16-row WMMA tiles)
#define LDS_PAD 132   // 132 % 64 == 4 -> conflict-free strided A-fragment reads

__device__ __forceinline__ void atom_add_f32(float* p, float v) {
  __hip_atomic_fetch_add(p, v, __ATOMIC_RELAXED, __HIP_MEMORY_SCOPE_AGENT);
}

// Async global->LDS 16B copy, tracked by ASYNCcnt (no VGPR round-trip).
__device__ __forceinline__ void async_g2l_b128(unsigned lds_addr,
                                               unsigned long long gaddr) {
  asm volatile("global_load_async_to_lds_b128 %0, %1, off"
               :
               : "v"(lds_addr), "v"(gaddr)
               : "memory");
}
__device__ __forceinline__ void wait_async0() {
  asm volatile("s_wait_asynccnt 0x0" ::: "memory");
}

// --------------------------------------------------------------------------
// One wave (32 lanes) per edge: lane l moves feature chunk [4l, 4l+4).
// Gather is a coalesced 512B read per edge; scatter is 4 fp32 L2 atomics/lane.
// --------------------------------------------------------------------------
__global__ __launch_bounds__(256) void sage_scatter(
    const float* __restrict__ hsrc, const int* __restrict__ src,
    const int* __restrict__ dst, float* __restrict__ nsum,
    float* __restrict__ deg, int E) {
  long long t = (long long)blockIdx.x * blockDim.x + threadIdx.x;
  int e = (int)(t >> 5);
  int lane = (int)(t & 31);
  if (e >= E) return;
  int s = src[e];
  int d = dst[e];
  const float4 v = *(const float4*)(hsrc + (size_t)s * D + lane * 4);
  float* b = nsum + (size_t)d * D + lane * 4;
  atom_add_f32(b + 0, v.x);
  atom_add_f32(b + 1, v.y);
  atom_add_f32(b + 2, v.z);
  atom_add_f32(b + 3, v.w);
  if (lane == 0) atom_add_f32(deg + d, 1.0f);
}

// --------------------------------------------------------------------------
// Fused per-relation dense update:
//   out[0:nt, 0:128] (accumulate or overwrite)
//     = hself @ Wself + (nsum/max(deg,1)) @ Wneigh + bias
//
// Block = 128 threads (4 waves) per 32-row stripe. The 32x128 self tile and
// raw nsum tile are staged into LDS with async b128 loads; the mean division
// is folded into the A-fragment read (per-lane scalar 1/max(deg,1), since a
// lane's A rows are fixed). Wave w owns N-tiles {2w, 2w+1}; each k-step does
// 8 WMMAs (2 row tiles x 2 N-tiles x {self, neigh}) against 8 B dword loads.
// Out-of-range rows are source-clamped (they only corrupt their own output
// row, which is store-masked), so EXEC stays all-1s for async + WMMA.
//
// fp32 WMMA fragment layouts (ISA 7.12.2):
//   A (16x4 MxK): lanes 0-15 -> M=lane, v0=K, v1=K+1; lanes 16-31 -> M=lane-16,
//                 v0=K+2, v1=K+3
//   B (4x16 KxN): lanes 0-15 -> N=lane, v0=row K, v1=row K+1; lanes 16-31 ->
//                 N=lane-16, v0=row K+2, v1=row K+3
//   C/D (16x16):  VGPR v: lanes 0-15 -> M=v, lanes 16-31 -> M=v+8; N=lane%16
// --------------------------------------------------------------------------
__global__ __launch_bounds__(128) void sage_gemm(
    const float* __restrict__ hself, const float* __restrict__ nsum,
    const float* __restrict__ deg, const float* __restrict__ Wself,
    const float* __restrict__ Wneigh, const float* __restrict__ bias,
    float* __restrict__ out, int nt, int accumulate) {
  __shared__ float sh[2][ROWS][LDS_PAD];

  const int tid = threadIdx.x;
  const int row0 = blockIdx.x * ROWS;

  // Async staging: 2 matrices x 32 rows x 128 cols = 2048 float4 slots,
  // 16 async b128 issues per thread.
#pragma unroll
  for (int i = 0; i < 8; ++i) {
    int flat = i * 128 + tid;        // 1024 float4 slots per matrix
    int r = flat >> 5;               // 32 float4 per row
    int c4 = (flat & 31) << 2;
    int gr = row0 + r;
    if (gr > nt - 1) gr = nt - 1;    // clamp: keeps EXEC full, rows masked later
    unsigned l0 = (unsigned)(unsigned long long)(&sh[0][r][c4]);
    unsigned l1 = (unsigned)(unsigned long long)(&sh[1][r][c4]);
    unsigned long long g0 =
        (unsigned long long)(hself + (size_t)gr * D + c4);
    unsigned long long g1 =
        (unsigned long long)(nsum + (size_t)gr * D + c4);
    async_g2l_b128(l0, g0);
    async_g2l_b128(l1, g1);
  }
  wait_async0();
  __syncthreads();

  const int lane = tid & 31;
  const int wave = tid >> 5;
  const int nlo = lane & 15;
  const int khalf = lane >> 4;  // 0: K,K+1   1: K+2,K+3
  const int koff = khalf * 2;

  // Per-lane mean scale for this lane's two A rows (nlo and 16+nlo).
  int rc0 = row0 + nlo;       if (rc0 > nt - 1) rc0 = nt - 1;
  int rc1 = row0 + 16 + nlo;  if (rc1 > nt - 1) rc1 = nt - 1;
  const float invd0 = 1.0f / fmaxf(deg[rc0], 1.0f);
  const float invd1 = 1.0f / fmaxf(deg[rc1], 1.0f);

  v8f acc_s[2][2], acc_n[2][2];  // [row tile][n tile]
#pragma unroll
  for (int n2 = 0; n2 < 2; ++n2) {
    const float bv = bias[(wave * 2 + n2) * 16 + nlo];
#pragma unroll
    for (int v = 0; v < 8; ++v) {
      acc_s[0][n2][v] = bv;  // seed bias into the self chains
      acc_s[1][n2][v] = bv;
      acc_n[0][n2][v] = 0.0f;
      acc_n[1][n2][v] = 0.0f;
    }
  }

#pragma unroll 4
  for (int k = 0; k < D; k += 4) {
    v2f aS0, aS1, aM0, aM1;
    aS0.x = sh[0][nlo][k + koff];
    aS0.y = sh[0][nlo][k + koff + 1];
    aS1.x = sh[0][16 + nlo][k + koff];
    aS1.y = sh[0][16 + nlo][k + koff + 1];
    aM0.x = sh[1][nlo][k + koff] * invd0;
    aM0.y = sh[1][nlo][k + koff + 1] * invd0;
    aM1.x = sh[1][16 + nlo][k + koff] * invd1;
    aM1.y = sh[1][16 + nlo][k + koff + 1] * invd1;
#pragma unroll
    for (int n2 = 0; n2 < 2; ++n2) {
      const int cb = (wave * 2 + n2) * 16 + nlo;
      v2f bS, bN;
      bS.x = Wself[(size_t)(k + koff) * D + cb];
      bS.y = Wself[(size_t)(k + koff + 1) * D + cb];
      bN.x = Wneigh[(size_t)(k + koff) * D + cb];
      bN.y = Wneigh[(size_t)(k + koff + 1) * D + cb];
      acc_s[0][n2] = __builtin_amdgcn_wmma_f32_16x16x4_f32(
          false, aS0, false, bS, (short)0, acc_s[0][n2], false, false);
      acc_s[1][n2] = __builtin_amdgcn_wmma_f32_16x16x4_f32(
          false, aS1, false, bS, (short)0, acc_s[1][n2], false, false);
      acc_n[0][n2] = __builtin_amdgcn_wmma_f32_16x16x4_f32(
          false, aM0, false, bN, (short)0, acc_n[0][n2], false, false);
      acc_n[1][n2] = __builtin_amdgcn_wmma_f32_16x16x4_f32(
          false, aM1, false, bN, (short)0, acc_n[1][n2], false, false);
    }
  }

#pragma unroll
  for (int rt = 0; rt < 2; ++rt) {
#pragma unroll
    for (int n2 = 0; n2 < 2; ++n2) {
      const int cb = (wave * 2 + n2) * 16 + nlo;
#pragma unroll
      for (int v = 0; v < 8; ++v) {
        int gr = row0 + rt * 16 + v + khalf * 8;
        if (gr < nt) {
          float* p = out + (size_t)gr * D + cb;
          float val = acc_s[rt][n2][v] + acc_n[rt][n2][v];
          if (accumulate) *p += val; else *p = val;
        }
      }
    }
  }
}

__global__ void relu_kernel(float* __restrict__ p, long long n) {
  long long i = (long long)blockIdx.x * blockDim.x + threadIdx.x;
  const long long stride = (long long)gridDim.x * blockDim.x;
  for (; i < n; i += stride) p[i] = fmaxf(p[i], 0.0f);
}

// ---------------------------------------------------------------------------
extern "C" void kernel_launch(void* const* d_in, const int* in_sizes, int n_in,
                              void* d_out, int out_size, void* d_ws,
                              size_t ws_size, hipStream_t stream) {
  (void)n_in; (void)out_size; (void)ws_size;

  const float* xq  = (const float*)d_in[0];
  const float* xp  = (const float*)d_in[1];
  const int*   src = (const int*)d_in[2];
  const int*   dst = (const int*)d_in[3];
  const float* Ws1 = (const float*)d_in[4];
  const float* Wn1 = (const float*)d_in[5];
  const float* b1  = (const float*)d_in[6];
  const float* Ws2 = (const float*)d_in[7];
  const float* Wn2 = (const float*)d_in[8];
  const float* b2  = (const float*)d_in[9];

  const int NQn = in_sizes[0] / D;
  const int NPn = in_sizes[1] / D;
  const int E   = in_sizes[2] / 4;
  const int NMAX = NQn > NPn ? NQn : NPn;

  // Workspace: layer-1 activations (contiguous so one ReLU pass covers both),
  // one reusable neighbor-sum buffer, one degree buffer.
  float* wsf  = (float*)d_ws;
  float* h1q  = wsf;
  float* h1p  = h1q + (size_t)NQn * D;
  float* nsum = h1p + (size_t)NPn * D;
  float* degb = nsum + (size_t)NMAX * D;

  float* oq = (float*)d_out;
  float* op = oq + (size_t)NQn * D;

  // relation r: (src_type, dst_type); 0=Query, 1=Product
  const int REL_S[4] = {0, 1, 1, 0};
  const int REL_T[4] = {1, 0, 0, 1};

  auto layer = [&](const float* hq, const float* hp, const float* Wself,
                   const float* Wneigh, const float* bias, float* outq,
                   float* outp) {
    int seen[2] = {0, 0};  // first relation per target overwrites out
    for (int r = 0; r < 4; ++r) {
      const int s = REL_S[r], t = REL_T[r];
      const float* hsrc  = (s == 0) ? hq : hp;
      const float* hself = (t == 0) ? hq : hp;
      float* outt = (t == 0) ? outq : outp;
      const int nt = (t == 0) ? NQn : NPn;

      hipMemsetAsync(nsum, 0, (size_t)nt * D * sizeof(float), stream);
      hipMemsetAsync(degb, 0, (size_t)nt * sizeof(float), stream);

      {
        long long work = (long long)E * 32;
        int blocks = (int)((work + 255) / 256);
        sage_scatter<<<blocks, 256, 0, stream>>>(
            hsrc, src + (size_t)r * E, dst + (size_t)r * E, nsum, degb, E);
      }
      {
        int blocks = (nt + ROWS - 1) / ROWS;
        sage_gemm<<<blocks, 128, 0, stream>>>(
            hself, nsum, degb, Wself + (size_t)r * D * D,
            Wneigh + (size_t)r * D * D, bias + (size_t)r * D, outt, nt,
            seen[t]);
      }
      seen[t] = 1;
    }
  };

  // Layer 1 -> h1 buffers, ReLU in place, Layer 2 -> d_out.
  layer(xq, xp, Ws1, Wn1, b1, h1q, h1p);
  relu_kernel<<<2048, 256, 0, stream>>>(h1q, (long long)(NQn + NPn) * D);
  layer(h1q, h1p, Ws2, Wn2, b2, oq, op);
}